// Quantizer_9818295239045
// MI455X (gfx1250) — compile-verified
//
#include <hip/hip_runtime.h>

typedef float v2f __attribute__((ext_vector_type(2)));
typedef float v4f __attribute__((ext_vector_type(4)));
typedef float v8f __attribute__((ext_vector_type(8)));
typedef unsigned int u32x4 __attribute__((ext_vector_type(4)));
typedef unsigned int u32x8 __attribute__((ext_vector_type(8)));

#define N_ROWS   65536
#define E_DIM    64
#define N_E      1024
#define HW       8192        /* H*W = 64*128 */
#define CHW      524288      /* C*H*W */
#define NUMEL    4194304     /* B*C*H*W */
#define BETA     0.25f
#define WAVES    8
#define NTILES   4096        /* N_ROWS / 16 */
#define CHUNK    64          /* codes per TDM chunk */
#define NCHUNK   16          /* N_E / CHUNK */
#define LDS_STRIDE 68        /* 64 + 4 pad DWORDs per row */

/* d_out layout (floats, concatenated in reference return order) */
#define OFF_LOSS ((size_t)0)
#define OFF_ZQ   ((size_t)1)
#define OFF_MC   ((size_t)1 + NUMEL)
#define OFF_MIDX (OFF_MC + (size_t)N_ROWS * N_E)
#define OFF_PERP (OFF_MIDX + N_ROWS)

/* workspace layout (floats) */
#define WS_LOSS  0      /* [4096] per-tile loss partials */
#define WS_NORM  4096   /* [1024] codebook squared norms */
#define WS_HIST  5120   /* [1024] uint histogram */

__global__ __launch_bounds__(1024) void vq_init(const float* __restrict__ cb,
                                                float* __restrict__ ws) {
    int t = threadIdx.x;  // 1024 threads, 1 block
    float s = 0.f;
    #pragma unroll 8
    for (int k = 0; k < E_DIM; ++k) {
        float v = cb[(size_t)t * E_DIM + k];
        s += v * v;
    }
    ws[WS_NORM + t] = s;
    ((unsigned int*)ws)[WS_HIST + t] = 0u;
}

// Tensor Data Mover: DMA one 64x64 fp32 codebook chunk (row stride 64) from
// global into LDS at lds_byte, padding 4 DWORDs after every 64 DWORDs so rows
// land at stride 68 floats (bank-conflict-free B fetches). D# per ISA 8.3/8.4.
__device__ __forceinline__ void tdm_load_chunk(const float* gsrc,
                                               unsigned lds_byte) {
    unsigned long long ga = (unsigned long long)gsrc;
    u32x4 g0;
    g0[0] = 1u;                                   // count=1 (valid), user mode
    g0[1] = lds_byte;                             // lds_addr
    g0[2] = (unsigned)(ga & 0xFFFFFFFFu);         // global_addr[31:0]
    g0[3] = (unsigned)((ga >> 32) & 0x1FFFFFFu)   // global_addr[56:32]
          | (2u << 30);                           // type=2 ("image")
    u32x8 g1;
    g1[0] = (2u << 16)    // data_size = 4B
          | (1u << 20)    // pad_enable
          | (5u << 22)    // pad_interval: 64 DWORDs
          | (3u << 25);   // pad_amount: 4 DWORDs
    g1[1] = (unsigned)E_DIM << 16;   // tensor_dim0[15:0] = 64
    g1[2] = (unsigned)CHUNK << 16;   // dim0 hi=0 | tensor_dim1[15:0] = 64
    g1[3] = (unsigned)E_DIM << 16;   // dim1 hi=0 | tile_dim0 = 64
    g1[4] = (unsigned)CHUNK;         // tile_dim1 = 64 | tile_dim2 = 0
    g1[5] = (unsigned)E_DIM;         // tensor_dim0_stride = 64
    g1[6] = 0u;                      // stride0 hi | tensor_dim1_stride lo (unused)
    g1[7] = 0u;
    u32x4 gz = {0u, 0u, 0u, 0u};     // groups 2/3: tile_dim2/3 = 0 -> unused
    asm volatile("tensor_load_to_lds %0, %1, %2, %3"
                 :: "s"(g0), "s"(g1), "s"(gz), "s"(gz)
                 : "memory");
}

__global__ __launch_bounds__(256) void vq_main(const float* __restrict__ z,
                                               const float* __restrict__ cb,
                                               float* __restrict__ out,
                                               float* __restrict__ ws) {
    __shared__ float cbT[2][CHUNK * LDS_STRIDE];  // double-buffered TDM target
    __shared__ float cbN[2][CHUNK];               // chunk norms
    __shared__ int   idxS[WAVES][16];             // per-wave argmin indices

    const int tid  = threadIdx.x;
    const int lane = tid & 31;
    const int wave = tid >> 5;
    const int l16  = lane & 15;
    const int lhi  = lane >> 4;

    const int tile = blockIdx.x * WAVES + wave;  // 0..4095
    const int n0   = tile * 16;
    const int bb   = n0 >> 13;                   // batch index (16 | 8192)
    const int hw0  = n0 & 8191;
    const size_t zoff0 = (size_t)bb * CHW + (size_t)hw0;  // + c*HW + r

    // ---- A: 16 rows x 64 K in WMMA 16x4 layout; row M = l16 ----
    const size_t arow = zoff0 + (size_t)l16;
    v2f a[16];
    #pragma unroll
    for (int kt = 0; kt < 16; ++kt) {
        int c0 = 4 * kt + 2 * lhi;
        a[kt].x = z[arow + (size_t)c0 * HW];
        a[kt].y = z[arow + (size_t)(c0 + 1) * HW];
    }

    float minv[8];
    int   mini[8];
    #pragma unroll
    for (int v = 0; v < 8; ++v) { minv[v] = 3.4e38f; mini[v] = 0; }

    const float* cbn_ws = ws + WS_NORM;
    const unsigned lds0 = (unsigned)(unsigned long long)&cbT[0][0];
    const unsigned lds1 = (unsigned)(unsigned long long)&cbT[1][0];

    // prime the pipeline: chunk 0 -> buffer 0 (TDM, wave 0 only; EXEC ignored)
    if (wave == 0) tdm_load_chunk(cb, lds0);

    for (int cc = 0; cc < NCHUNK; ++cc) {
        const int buf = cc & 1;
        if (wave == 0) {
            if (cc + 1 < NCHUNK) {   // prefetch next chunk into other buffer
                tdm_load_chunk(cb + (size_t)(cc + 1) * CHUNK * E_DIM,
                               buf ? lds0 : lds1);
                __builtin_amdgcn_s_wait_tensorcnt(1);  // chunk cc landed
            } else {
                __builtin_amdgcn_s_wait_tensorcnt(0);
            }
        }
        if (tid < CHUNK) cbN[buf][tid] = cbn_ws[cc * CHUNK + tid];
        __syncthreads();  // chunk cc + norms visible to all waves

        #pragma unroll
        for (int en0 = 0; en0 < CHUNK; en0 += 16) {
            // batch all B fragments first -> staggered dscnt waits + WMMA chain
            const float* bp = &cbT[buf][(en0 + l16) * LDS_STRIDE + 2 * lhi];
            v2f bv[16];
            #pragma unroll
            for (int kt = 0; kt < 16; ++kt)
                bv[kt] = *(const v2f*)(bp + 4 * kt);   // ds_load_2addr_b64 x8
            v8f c = {0.f, 0.f, 0.f, 0.f, 0.f, 0.f, 0.f, 0.f};
            #pragma unroll
            for (int kt = 0; kt < 16; ++kt)
                c = __builtin_amdgcn_wmma_f32_16x16x4_f32(
                        false, a[kt], false, bv[kt], (short)0, c, false, false);
            int   code = cc * CHUNK + en0 + l16;
            float nrm  = cbN[buf][en0 + l16];
            #pragma unroll
            for (int v = 0; v < 8; ++v) {
                float m = nrm - 2.0f * c[v];   // ||e||^2 - 2 z.e
                if (m < minv[v]) { minv[v] = m; mini[v] = code; }
            }
        }
        __syncthreads();  // release buffer `buf` for reuse at iteration cc+2
    }

    // cross-lane (min,argmin) reduce within each 16-lane half (codes across lanes)
    #pragma unroll
    for (int v = 0; v < 8; ++v) {
        float bv = minv[v];
        int   bi = mini[v];
        #pragma unroll
        for (int m = 8; m >= 1; m >>= 1) {
            float ov = __shfl_xor(bv, m, 32);
            int   oi = __shfl_xor(bi, m, 32);
            if (ov < bv || (ov == bv && oi < bi)) { bv = ov; bi = oi; }
        }
        minv[v] = bv; mini[v] = bi;
    }
    if (l16 == 0) {
        #pragma unroll
        for (int v = 0; v < 8; ++v) idxS[wave][v + 8 * lhi] = mini[v];
    }
    __syncthreads();

    // ---- min_index (as float) ----
    if (lane < 16)
        out[OFF_MIDX + (size_t)n0 + lane] = (float)idxS[wave][lane];

    // ---- one-hot min_codes: 256 MB write-once stream -> non-temporal stores
    //      (L2 is 192 MB; don't let this stream evict codebook/z lines) ----
    {
        float* mc = out + OFF_MC + (size_t)n0 * N_E;
        for (int r = 0; r < 16; ++r) {
            int    idx  = idxS[wave][r];
            float* rowp = mc + (size_t)r * N_E;
            #pragma unroll 8
            for (int j = 0; j < 32; ++j) {
                int e = j * 32 + lane;
                __builtin_nontemporal_store((e == idx) ? 1.0f : 0.0f, rowp + e);
            }
        }
    }

    // ---- z_q (B,C,H,W layout) + loss partial ----
    float acc = 0.f;
    {
        int r   = l16;
        int idx = idxS[wave][r];
        const float* cbrow = cb + (size_t)idx * E_DIM;
        #pragma unroll 4
        for (int cit = 0; cit < 32; ++cit) {
            int    c   = 2 * cit + lhi;
            size_t off = zoff0 + (size_t)c * HW + r;
            float  cbv = cbrow[c];
            float  zv  = __builtin_nontemporal_load(z + off);  // last use of z
            __builtin_nontemporal_store(cbv, out + OFF_ZQ + off);  // z_q_st==z_q
            float d = zv - cbv;
            acc += d * d;
        }
    }
    #pragma unroll
    for (int m = 16; m >= 1; m >>= 1) acc += __shfl_xor(acc, m, 32);
    if (lane == 0) ws[WS_LOSS + tile] = acc;                 // deterministic
    if (lane < 16)
        atomicAdd(&((unsigned int*)ws)[WS_HIST + idxS[wave][lane]], 1u);
}

__global__ __launch_bounds__(1024) void vq_fin(const float* __restrict__ ws,
                                               float* __restrict__ out) {
    __shared__ float redL[1024];
    __shared__ float redE[1024];
    int t = threadIdx.x;

    float ls = ws[WS_LOSS + t] + ws[WS_LOSS + t + 1024] +
               ws[WS_LOSS + t + 2048] + ws[WS_LOSS + t + 3072];
    redL[t] = ls;

    const unsigned int* hist = (const unsigned int*)ws + WS_HIST;
    float em = (float)hist[t] * (1.0f / (float)N_ROWS);
    redE[t] = em * logf(em + 1e-10f);
    __syncthreads();

    for (int s = 512; s > 0; s >>= 1) {
        if (t < s) { redL[t] += redL[t + s]; redE[t] += redE[t + s]; }
        __syncthreads();
    }
    if (t == 0) {
        out[OFF_LOSS] = redL[0] * ((1.0f + BETA) / (float)NUMEL);
        out[OFF_PERP] = expf(-redE[0]);
    }
}

extern "C" void kernel_launch(void* const* d_in, const int* in_sizes, int n_in,
                              void* d_out, int out_size, void* d_ws, size_t ws_size,
                              hipStream_t stream) {
    const float* z  = (const float*)d_in[0];
    const float* cb = (const float*)d_in[1];
    float* out = (float*)d_out;
    float* ws  = (float*)d_ws;

    vq_init<<<1, 1024, 0, stream>>>(cb, ws);
    vq_main<<<NTILES / WAVES, 256, 0, stream>>>(z, cb, out, ws);
    vq_fin<<<1, 1024, 0, stream>>>(ws, out);
}